// AE_rnn_57002805953276
// MI455X (gfx1250) — compile-verified
//
#include <hip/hip_runtime.h>
#include <cstddef>

#define L_SEQ 2048
#define VOCAB 50257
#define NPAD  50272           // VOCAB rounded up to 16
#define KPAD  28              // K=26 rounded up to 4 (7 WMMA k-steps)
#define HID 13
#define GATES 52              // 4*HID
#define IN_DIM 256

typedef float v8f __attribute__((ext_vector_type(8)));
typedef float v2f __attribute__((ext_vector_type(2)));

__device__ __forceinline__ float sigf(float x) { return 1.0f / (1.0f + __expf(-x)); }

// ---------------------------------------------------------------- embedding
__global__ void embed_kernel(const int* __restrict__ x,
                             const float* __restrict__ embW,
                             float* __restrict__ embs) {
    int idx = blockIdx.x * blockDim.x + threadIdx.x;   // < L_SEQ*IN_DIM
    int t = idx >> 8, j = idx & 255;
    embs[idx] = embW[(size_t)x[t] * IN_DIM + j];
}

// ------------------------------------------- input-side GEMM: pre = Wih@x + b
__global__ void pregemm_kernel(const float* __restrict__ in,    // [L][K]
                               const float* __restrict__ Wih,   // [2][52][K]
                               const float* __restrict__ b,     // [2][52]
                               float* __restrict__ pre,         // [2][L][52]
                               int K) {
    __shared__ float xs[IN_DIM];
    int bid = blockIdx.x;            // 0 .. 2*L-1
    int d = bid >> 11;
    int t = bid & (L_SEQ - 1);
    for (int k = threadIdx.x; k < K; k += 64) xs[k] = in[(size_t)t * K + k];
    __syncthreads();
    int g = threadIdx.x;
    if (g < GATES) {
        float acc = b[d * GATES + g];
        const float* w = Wih + ((size_t)d * GATES + g) * K;
        for (int k = 0; k < K; ++k) acc += w[k] * xs[k];
        pre[((size_t)d * L_SEQ + t) * GATES + g] = acc;
    }
}

// -------------------------------------------------- sequential LSTM scan
__global__ void lstm_scan_kernel(const float* __restrict__ pre,   // [2][L][52]
                                 const float* __restrict__ Whh,   // [2][52][13]
                                 float* __restrict__ hs,          // [L][26] (if writeHs)
                                 float* __restrict__ finals,      // [104]
                                 int layer, int writeHs) {
    const int d = blockIdx.x;
    const int g = threadIdx.x;
    __shared__ float h[HID], c[HID], z[GATES];
    float whh[HID];
    if (g < GATES)
        for (int i = 0; i < HID; ++i) whh[i] = Whh[((size_t)d * GATES + g) * HID + i];
    if (g < HID) { h[g] = 0.0f; c[g] = 0.0f; }
    __syncthreads();
    for (int s = 0; s < L_SEQ; ++s) {
        int t = (d == 1) ? (L_SEQ - 1 - s) : s;
        if (g < GATES) {
            float acc = pre[((size_t)d * L_SEQ + t) * GATES + g];
            #pragma unroll
            for (int i = 0; i < HID; ++i) acc += whh[i] * h[i];
            z[g] = acc;
        }
        __syncthreads();
        if (g < HID) {
            float zi = z[g], zf = z[HID + g], zg = z[2 * HID + g], zo = z[3 * HID + g];
            float c2 = sigf(zf) * c[g] + sigf(zi) * tanhf(zg);
            float h2 = sigf(zo) * tanhf(c2);
            c[g] = c2; h[g] = h2;
            if (writeHs) hs[(size_t)t * (2 * HID) + d * HID + g] = h2;
        }
        __syncthreads();
    }
    if (g < HID) {
        finals[(layer * 2 + d) * HID + g] = h[g];          // h_flat order: f0,b0,f1,b1
        finals[GATES + (layer * 2 + d) * HID + g] = c[g];  // c_flat at +52
    }
}

// ------------------------------- state projections + decoder step 0 (input -1)
// writes dec row 0 with padded stride KPAD (cols 26,27 zeroed)
__global__ void proj_dec0_kernel(const float* __restrict__ finals,
                                 const float* __restrict__ p1W, const float* __restrict__ p1b,
                                 const float* __restrict__ p2W, const float* __restrict__ p2b,
                                 const float* __restrict__ dWih, const float* __restrict__ dWhh,
                                 const float* __restrict__ db,
                                 float* __restrict__ dec_out) {
    __shared__ float sh[2 * HID], sc[2 * HID];
    int j = threadIdx.x;
    if (j < 2 * HID) {
        float ah = p1b[j], ac = p2b[j];
        for (int k = 0; k < GATES; ++k) {
            ah += p1W[j * GATES + k] * finals[k];
            ac += p2W[j * GATES + k] * finals[GATES + k];
        }
        sh[j] = ah; sc[j] = ac;
    }
    __syncthreads();
    if (j < 2 * HID) {
        int d = j / HID, g = j % HID;
        float zq[4];
        #pragma unroll
        for (int q = 0; q < 4; ++q) {
            int r = q * HID + g;
            float acc = dWih[d * GATES + r] * (-1.0f) + db[d * GATES + r];
            for (int k = 0; k < HID; ++k)
                acc += dWhh[((size_t)d * GATES + r) * HID + k] * sh[d * HID + k];
            zq[q] = acc;
        }
        float c2 = sigf(zq[1]) * sc[j] + sigf(zq[0]) * tanhf(zq[2]);
        dec_out[j] = sigf(zq[3]) * tanhf(c2);
    } else if (j < KPAD) {
        dec_out[j] = 0.0f;                 // K padding for row 0
    }
}

// ------------- decoder steps 1..L-1: zero initial state -> fully parallel
// padded stride KPAD; pad columns written as 0
__global__ void dec_rest_kernel(const int* __restrict__ x,
                                const float* __restrict__ dWih,
                                const float* __restrict__ db,
                                float* __restrict__ dec_out) {
    int idx = blockIdx.x * blockDim.x + threadIdx.x;
    if (idx >= (L_SEQ - 1) * KPAD) return;
    int t = 1 + idx / KPAD;
    int j = idx % KPAD;
    if (j >= 2 * HID) { dec_out[(size_t)t * KPAD + j] = 0.0f; return; }
    int d = j / HID, g = j % HID;
    float v = (float)x[t - 1];
    // c=0, h=0: sig(f)*c == 0, Whh@h == 0
    float zi = dWih[d * GATES + g]            * v + db[d * GATES + g];
    float zg = dWih[d * GATES + 2 * HID + g]  * v + db[d * GATES + 2 * HID + g];
    float zo = dWih[d * GATES + 3 * HID + g]  * v + db[d * GATES + 3 * HID + g];
    float c2 = sigf(zi) * tanhf(zg);
    dec_out[(size_t)t * KPAD + j] = sigf(zo) * tanhf(c2);
}

// -------------- pack out_W[50257][26] -> wP[50272][28] and out_b -> biasP[50272]
// zero-padded so the GEMM loads are unconditional
__global__ void pack_w_kernel(const float* __restrict__ W,
                              const float* __restrict__ bias,
                              float* __restrict__ wP,
                              float* __restrict__ biasP) {
    size_t idx = (size_t)blockIdx.x * blockDim.x + threadIdx.x;
    const size_t wtotal = (size_t)NPAD * KPAD;
    if (idx < wtotal) {
        int r = (int)(idx / KPAD), k = (int)(idx % KPAD);
        wP[idx] = (r < VOCAB && k < 26) ? W[(size_t)r * 26 + k] : 0.0f;
    } else if (idx < wtotal + NPAD) {
        int r = (int)(idx - wtotal);
        biasP[r] = (r < VOCAB) ? bias[r] : 0.0f;
    }
}

// -------------------- preds = dec(2048xKPAD) @ wP^T(KPADxNPAD) + biasP
// fp32 WMMA 16x16x4; 7 k-steps; all loads unconditional b64; NT stores.
__global__ void out_gemm_wmma(const float* __restrict__ decP,  // [2048][KPAD]
                              const float* __restrict__ wP,    // [NPAD][KPAD]
                              const float* __restrict__ biasP, // [NPAD]
                              float* __restrict__ preds) {     // [2048][VOCAB]
    const int lane = threadIdx.x;        // 0..31 (wave32)
    const int half = lane >> 4;          // K sub-pair select (ISA A/B layout)
    const int lm   = lane & 15;
    const int n0 = blockIdx.x * 16;
    const int m0 = blockIdx.y * 16;
    const int col = n0 + lm;

    const float bv = biasP[col];
    v8f acc;
    #pragma unroll
    for (int r = 0; r < 8; ++r) acc[r] = bv;   // fold bias into accumulator

    const v2f* Arow = (const v2f*)(decP + (size_t)(m0 + lm) * KPAD);
    const v2f* Brow = (const v2f*)(wP   + (size_t)col       * KPAD);
    #pragma unroll
    for (int kk = 0; kk < 7; ++kk) {
        v2f a = Arow[kk * 2 + half];           // K pair {k0,k0+1}, k0=kk*4+half*2
        v2f b = Brow[kk * 2 + half];
        acc = __builtin_amdgcn_wmma_f32_16x16x4_f32(
                  false, a, false, b, (short)0, acc, false, false);
    }

    const bool cok = (col < VOCAB);            // only last N-tile is partial
    #pragma unroll
    for (int r = 0; r < 8; ++r) {
        int row = m0 + r + half * 8;           // C/D layout: rows m0+r / m0+8+r
        if (cok)                                // streaming output: keep W in L2
            __builtin_nontemporal_store(acc[r], &preds[(size_t)row * VOCAB + col]);
    }
}

// ----------------------------------------------------------------- launcher
extern "C" void kernel_launch(void* const* d_in, const int* in_sizes, int n_in,
                              void* d_out, int out_size, void* d_ws, size_t ws_size,
                              hipStream_t stream) {
    const int*   x     = (const int*)  d_in[0];
    const float* embW  = (const float*)d_in[1];
    const float* Wih0  = (const float*)d_in[2];
    const float* Whh0  = (const float*)d_in[3];
    const float* b0    = (const float*)d_in[4];
    const float* Wih1  = (const float*)d_in[5];
    const float* Whh1  = (const float*)d_in[6];
    const float* b1    = (const float*)d_in[7];
    const float* p1W   = (const float*)d_in[8];
    const float* p1b   = (const float*)d_in[9];
    const float* p2W   = (const float*)d_in[10];
    const float* p2b   = (const float*)d_in[11];
    const float* dWih  = (const float*)d_in[12];
    const float* dWhh  = (const float*)d_in[13];
    const float* db    = (const float*)d_in[14];
    const float* outW  = (const float*)d_in[15];
    const float* outb  = (const float*)d_in[16];

    float* preds = (float*)d_out;                       // [2048][50257]
    float* embs  = preds + (size_t)L_SEQ * VOCAB;       // [2048][256]

    float* ws     = (float*)d_ws;
    float* pre0   = ws;                                  // 2*2048*52
    float* pre1   = pre0 + 2 * L_SEQ * GATES;            // 2*2048*52
    float* out0   = pre1 + 2 * L_SEQ * GATES;            // 2048*26
    float* decP   = out0 + L_SEQ * 2 * HID;              // 2048*28 (padded)
    float* finals = decP + L_SEQ * KPAD;                 // 104
    float* wP     = finals + 104;                        // 50272*28 (padded W)
    float* biasP  = wP + (size_t)NPAD * KPAD;            // 50272

    // 0) pack W/bias into zero-padded tiles (runs early, independent)
    {
        size_t total = (size_t)NPAD * KPAD + NPAD;
        pack_w_kernel<<<(unsigned)((total + 255) / 256), 256, 0, stream>>>(
            outW, outb, wP, biasP);
    }
    // 1) embedding (also output #2)
    embed_kernel<<<(L_SEQ * IN_DIM) / 256, 256, 0, stream>>>(x, embW, embs);
    // 2) encoder layer 0: input GEMM then bidirectional scan
    pregemm_kernel<<<2 * L_SEQ, 64, 0, stream>>>(embs, Wih0, b0, pre0, IN_DIM);
    lstm_scan_kernel<<<2, 64, 0, stream>>>(pre0, Whh0, out0, finals, 0, 1);
    // 3) encoder layer 1
    pregemm_kernel<<<2 * L_SEQ, 64, 0, stream>>>(out0, Wih1, b1, pre1, 2 * HID);
    lstm_scan_kernel<<<2, 64, 0, stream>>>(pre1, Whh1, out0, finals, 1, 0);
    // 4) projections + decoder step 0
    proj_dec0_kernel<<<1, 32, 0, stream>>>(finals, p1W, p1b, p2W, p2b,
                                           dWih, dWhh, db, decP);
    // 5) decoder steps 1..L-1 (zero-state, parallel)
    dec_rest_kernel<<<((L_SEQ - 1) * KPAD + 127) / 128, 128, 0, stream>>>(
        x, dWih, db, decP);
    // 6) big output GEMM via fp32 WMMA (store-bandwidth bound: ~412 MB)
    dim3 gg(NPAD / 16, L_SEQ / 16);
    out_gemm_wmma<<<gg, 32, 0, stream>>>(decP, wP, biasP, preds);
}